// DEAM_79044578116356
// MI455X (gfx1250) — compile-verified
//
#include <hip/hip_runtime.h>
#include <hip/hip_bf16.h>

// DEAM fused pipeline for MI455X (gfx1250), fp32 WMMA path.
//
// Sizes: B=8, C=64, H=W=128, P=16384 pixels/batch.
// ws layout (floats): [0, 25165824) em (q|k|v per batch, [b][o<192][p])
//                     [25165824, +12288) folded conv1 weights W2 [192][64]
//                     [then +192)        folded conv1 bias2
// ws bytes required ≈ 100.7 MB. em (96MB) is L2-resident (192MB L2).

typedef __attribute__((ext_vector_type(2))) float v2f;
typedef __attribute__((ext_vector_type(8))) float v8f;

#define EM_ELEMS   (8 * 192 * 16384)      // 25165824
#define W2_OFF     EM_ELEMS
#define B2_OFF     (EM_ELEMS + 192 * 64)

// ---------------------------------------------------------------- kernel 0
// Fold BatchNorm (eval stats) into 1x1-conv weights + bias.
__global__ __launch_bounds__(256) void deam_fold_bn(
    const float* __restrict__ w, const float* __restrict__ cb,
    const float* __restrict__ gamma, const float* __restrict__ beta,
    const float* __restrict__ mean, const float* __restrict__ var,
    float* __restrict__ w2, float* __restrict__ b2) {
  int idx = blockIdx.x * 256 + threadIdx.x;           // 192*64 = 12288
  if (idx >= 192 * 64) return;
  int o = idx >> 6;
  float inv = gamma[o] * rsqrtf(var[o] + 1e-5f);
  w2[idx] = w[idx] * inv;
  if ((idx & 63) == 0) b2[o] = cb[o] * inv + beta[o] - mean[o] * inv;
}

// ---------------------------------------------------------------- kernel 1
// em[b][o][p] = relu(W2 @ e + b2), o in [0,192).  One wave per 16-pixel
// N-tile; 12 M-tiles x 16 K-steps of v_wmma_f32_16x16x4_f32.
// WMMA f32 layouts (ISA 7.12.2):
//   A 16x4:  lane<16 -> row=lane, VGPR{0,1}=K{0,1}; lane>=16 -> K{2,3}
//   B 4x16:  VGPR0: K=0 (lanes 0-15), K=2 (lanes 16-31); VGPR1: K=1 / K=3
//   C/D:     VGPR r: M=r (lanes 0-15), M=r+8 (lanes 16-31), N=lane&15
__global__ __launch_bounds__(256) void deam_qkv_gemm(
    const float* __restrict__ e, const float* __restrict__ w2,
    const float* __restrict__ b2, float* __restrict__ em) {
  const int t    = threadIdx.x;
  const int lane = t & 31;
  const int wv   = t >> 5;
  const int nt   = blockIdx.x * 8 + wv;               // 0..8191
  const int b    = nt >> 10;
  const int p0   = (nt & 1023) << 4;
  const int l16  = lane & 15;
  const int hiOf = (lane < 16) ? 0 : 2;
  const int addM = (lane < 16) ? 0 : 8;

  // Preload B panel: e[4j+hiOf .. +1][p0+l16] for j=0..15 (32 VGPRs).
  const float* ebase = e + (size_t)b * 64 * 16384 + p0 + l16;
  v2f Bp[16];
#pragma unroll
  for (int j = 0; j < 16; ++j) {
    Bp[j].x = ebase[(size_t)(4 * j + hiOf) * 16384];
    Bp[j].y = ebase[(size_t)(4 * j + hiOf + 1) * 16384];
  }

#pragma unroll 2
  for (int mt = 0; mt < 12; ++mt) {
    const int m0 = mt << 4;
    const float* arow = w2 + (m0 + l16) * 64 + hiOf;
    v8f acc = {};
#pragma unroll
    for (int j = 0; j < 16; ++j) {
      v2f a;
      a.x = arow[4 * j];
      a.y = arow[4 * j + 1];
      acc = __builtin_amdgcn_wmma_f32_16x16x4_f32(
          false, a, false, Bp[j], (short)0, acc, false, false);
    }
#pragma unroll
    for (int r = 0; r < 8; ++r) {
      const int o = m0 + r + addM;
      float y = acc[r] + b2[o];
      y = fmaxf(y, 0.0f);
      em[((size_t)b * 192 + o) * 16384 + p0 + l16] = y;
    }
  }
}

// ---------------------------------------------------------------- kernel 2
// Per block: one 16x16 spatial tile of one batch.  Dilated-3x3 (dil=2)
// local attention via LDS-staged k/v halo chunks, then fc GEMM via WMMA.
__global__ __launch_bounds__(256) void deam_attn_fc(
    const float* __restrict__ em, const float* __restrict__ f,
    const float* __restrict__ fcw, const float* __restrict__ fcb,
    float* __restrict__ out) {
  extern __shared__ float lds[];
  float* kvbuf = lds;          // 16 ch x 400 halo px = 6400 f
  float* tbuf  = lds + 6400;   // 64 ch x 256 px      = 16384 f

  const int t    = threadIdx.x;
  const int lane = t & 31;
  const int wv   = t >> 5;
  const int blk  = blockIdx.x;           // 512 = 8 batches * 64 tiles
  const int b    = blk >> 6;
  const int tile = blk & 63;
  const int h0   = (tile >> 3) << 4;
  const int w0   = (tile & 7) << 4;
  const int ty   = t >> 4, tx = t & 15;
  const int pix  = (h0 + ty) * 128 + (w0 + tx);

  // halo indices of the 9 dilated neighbors of this thread's pixel
  int hp[9];
#pragma unroll
  for (int iy = 0; iy < 3; ++iy)
#pragma unroll
    for (int ix = 0; ix < 3; ++ix)
      hp[iy * 3 + ix] = (ty + 2 * iy) * 20 + (tx + 2 * ix);

  const size_t embase = (size_t)b * 192 * 16384;
  const float* qptr = em + embase;
  const float* kptr = em + embase + (size_t)64 * 16384;
  const float* vptr = em + embase + (size_t)128 * 16384;

  // ---- phase 1: logits over 4 channel chunks
  float lg[9] = {0.f, 0.f, 0.f, 0.f, 0.f, 0.f, 0.f, 0.f, 0.f};
  for (int cc = 0; cc < 4; ++cc) {
    const int c0 = cc << 4;
    __builtin_prefetch(kptr + (size_t)c0 * 16384 + pix, 0, 0);
    __syncthreads();
#pragma unroll
    for (int i = 0; i < 25; ++i) {               // 25*256 = 6400
      const int idx = i * 256 + t;
      const int c = idx / 400, pp = idx - c * 400;
      const int hy = pp / 20, hx = pp - hy * 20;
      const int gh = h0 - 2 + hy, gw = w0 - 2 + hx;
      float val = 0.f;
      if ((unsigned)gh < 128u && (unsigned)gw < 128u)
        val = kptr[(size_t)(c0 + c) * 16384 + gh * 128 + gw];
      kvbuf[idx] = val;
    }
    __syncthreads();
#pragma unroll
    for (int c = 0; c < 16; ++c) {
      const float qc = qptr[(size_t)(c0 + c) * 16384 + pix];
      const float* kr = kvbuf + c * 400;
#pragma unroll
      for (int n = 0; n < 9; ++n) lg[n] = fmaf(qc, kr[hp[n]], lg[n]);
    }
  }

  // ---- softmax over 9 neighbors, scale 1/sqrt(64)
  float mx = -3.4e38f;
#pragma unroll
  for (int n = 0; n < 9; ++n) { lg[n] *= 0.125f; mx = fmaxf(mx, lg[n]); }
  float sum = 0.f;
#pragma unroll
  for (int n = 0; n < 9; ++n) { lg[n] = __expf(lg[n] - mx); sum += lg[n]; }
  const float rs = 1.0f / sum;
#pragma unroll
  for (int n = 0; n < 9; ++n) lg[n] *= rs;

  // ---- phase 2: out1 + f_map into tbuf (64 ch x 256 px)
  for (int cc = 0; cc < 4; ++cc) {
    const int c0 = cc << 4;
    __builtin_prefetch(vptr + (size_t)c0 * 16384 + pix, 0, 0);
    __syncthreads();
#pragma unroll
    for (int i = 0; i < 25; ++i) {
      const int idx = i * 256 + t;
      const int c = idx / 400, pp = idx - c * 400;
      const int hy = pp / 20, hx = pp - hy * 20;
      const int gh = h0 - 2 + hy, gw = w0 - 2 + hx;
      float val = 0.f;
      if ((unsigned)gh < 128u && (unsigned)gw < 128u)
        val = vptr[(size_t)(c0 + c) * 16384 + gh * 128 + gw];
      kvbuf[idx] = val;
    }
    __syncthreads();
#pragma unroll
    for (int c = 0; c < 16; ++c) {
      const float* vr = kvbuf + c * 400;
      float s = 0.f;
#pragma unroll
      for (int n = 0; n < 9; ++n) s = fmaf(lg[n], vr[hp[n]], s);
      s += f[((size_t)b * 64 + c0 + c) * 16384 + pix];
      tbuf[(c0 + c) * 256 + t] = s;
    }
  }
  __syncthreads();

  // ---- phase 3: fc GEMM (64x64 @ 64x256) via WMMA f32 16x16x4 from LDS
  const int l16  = lane & 15;
  const int hiOf = (lane < 16) ? 0 : 2;
  const int addM = (lane < 16) ? 0 : 8;
#pragma unroll
  for (int i = 0; i < 8; ++i) {            // 8 waves x 8 tiles = 4M x 16N
    const int tid8 = wv * 8 + i;
    const int m0 = (tid8 & 3) << 4;
    const int p0 = (tid8 >> 2) << 4;
    const float* arow = fcw + (m0 + l16) * 64 + hiOf;
    const float* brow = tbuf + p0 + l16;
    v8f acc = {};
#pragma unroll
    for (int j = 0; j < 16; ++j) {
      v2f a, bb;
      a.x  = arow[4 * j];
      a.y  = arow[4 * j + 1];
      bb.x = brow[(4 * j + hiOf) * 256];
      bb.y = brow[(4 * j + hiOf + 1) * 256];
      acc = __builtin_amdgcn_wmma_f32_16x16x4_f32(
          false, a, false, bb, (short)0, acc, false, false);
    }
#pragma unroll
    for (int r = 0; r < 8; ++r) {
      const int o  = m0 + r + addM;
      const int pl = p0 + l16;
      const int gy = h0 + (pl >> 4), gx = w0 + (pl & 15);
      out[((size_t)b * 64 + o) * 16384 + gy * 128 + gx] = acc[r] + fcb[o];
    }
  }
}

// ---------------------------------------------------------------- launch
extern "C" void kernel_launch(void* const* d_in, const int* in_sizes, int n_in,
                              void* d_out, int out_size, void* d_ws, size_t ws_size,
                              hipStream_t stream) {
  const float* f_map   = (const float*)d_in[0];
  const float* e_map   = (const float*)d_in[1];
  const float* conv1_w = (const float*)d_in[2];
  const float* conv1_b = (const float*)d_in[3];
  const float* bn_g    = (const float*)d_in[4];
  const float* bn_b    = (const float*)d_in[5];
  const float* bn_m    = (const float*)d_in[6];
  const float* bn_v    = (const float*)d_in[7];
  const float* fc_w    = (const float*)d_in[8];
  const float* fc_b    = (const float*)d_in[9];
  float* outp = (float*)d_out;

  float* ws   = (float*)d_ws;
  float* em   = ws;               // 96 MB intermediate, L2-resident
  float* w2   = ws + W2_OFF;
  float* b2   = ws + B2_OFF;

  deam_fold_bn<<<48, 256, 0, stream>>>(conv1_w, conv1_b, bn_g, bn_b, bn_m,
                                       bn_v, w2, b2);
  deam_qkv_gemm<<<1024, 256, 0, stream>>>(e_map, w2, b2, em);
  const size_t shbytes = (6400 + 16384) * sizeof(float);   // 91 KB LDS
  deam_attn_fc<<<512, 256, shbytes, stream>>>(em, f_map, fc_w, fc_b, outp);
}